// BiLSTMCRF_8349416423715
// MI455X (gfx1250) — compile-verified
//
#include <hip/hip_runtime.h>

#define B_    64
#define S_    512
#define EMB_  128
#define HID_  256
#define G4_   1024       // 4*HID
#define NL_   9
#define M_    (B_*S_)    // 32768
#define HSTR  260        // LDS row stride (== 4 mod 64 -> conflict-free b64 frag loads)

typedef float v2f __attribute__((ext_vector_type(2)));
typedef float v8f __attribute__((ext_vector_type(8)));

__device__ __forceinline__ v8f wmma_f32(v2f a, v2f b, v8f c) {
    // D = A(16x4) * B(4x16) + C, full fp32 (CDNA5 V_WMMA_F32_16X16X4_F32)
    return __builtin_amdgcn_wmma_f32_16x16x4_f32(false, a, false, b, (short)0, c, false, false);
}

// Branch-free fast transcendentals on the hardware TRANS pipe
__device__ __forceinline__ float fast_sigmoid(float x) {
    return __builtin_amdgcn_rcpf(1.0f + __expf(-x));
}
__device__ __forceinline__ float fast_tanh(float x) {
#if __has_builtin(__builtin_amdgcn_tanhf)
    return __builtin_amdgcn_tanhf(x);          // gfx1250 v_tanh_f32
#elif __has_builtin(__builtin_amdgcn_tanh_f32)
    return __builtin_amdgcn_tanh_f32(x);
#else
    // 1 - 2/(e^{2x}+1): correct limits at +/-inf, branch-free
    return 1.0f - 2.0f * __builtin_amdgcn_rcpf(__expf(2.0f * x) + 1.0f);
#endif
}

// ---------------- embedding gather ----------------
__global__ __launch_bounds__(128)
void embed_kernel(const int* __restrict__ tokens, const float* __restrict__ emb,
                  float* __restrict__ X0) {
    int bs = blockIdx.x;                       // 0..M_-1
    int tok = tokens[bs];
    X0[(size_t)bs * EMB_ + threadIdx.x] = emb[(size_t)tok * EMB_ + threadIdx.x];
}

// ---------------- Whh (1024 x 256) -> Wt (256 x 1024), K-major for coalesced B frags --
__global__ __launch_bounds__(256)
void transpose_kernel(const float* __restrict__ W, float* __restrict__ Wt) {
    int idx = blockIdx.x * 256 + threadIdx.x;  // over 256*1024 outputs
    int n = idx & (G4_ - 1);
    int k = idx >> 10;
    Wt[(size_t)k * G4_ + n] = W[(size_t)n * HID_ + k];
}

// ---------------- gx = X(M x K) * W^T(K x 1024) + (bih+bhh) ----------------
// grid: (M/64, 1024/128), block 256 (8 waves: 4 M-waves x 2 N-waves)
template<int K>
__global__ __launch_bounds__(256)
void gx_gemm_kernel(const float* __restrict__ X, const float* __restrict__ W,
                    const float* __restrict__ bih, const float* __restrict__ bhh,
                    float* __restrict__ G) {
    const int tid  = threadIdx.x;
    const int wave = tid >> 5, lane = tid & 31;
    const int li   = lane & 15, half = lane >> 4;
    const int wm   = wave & 3,  wn   = wave >> 2;
    const int mBase = blockIdx.x * 64 + wm * 16;
    const int nBase = blockIdx.y * 128 + wn * 64;      // this wave: 4 N-tiles of 16

    v8f acc[4] = {};
    const float* Arow = X + (size_t)(mBase + li) * K + 2 * half;
    const float* Brow = W + (size_t)(nBase + li) * K + 2 * half;

    for (int k = 0; k < K; k += 4) {
        v2f a; a.x = Arow[k]; a.y = Arow[k + 1];
        #pragma unroll
        for (int t = 0; t < 4; ++t) {
            const float* Bp = Brow + (size_t)(t * 16) * K + k;
            v2f b; b.x = Bp[0]; b.y = Bp[1];
            acc[t] = wmma_f32(a, b, acc[t]);
        }
    }
    #pragma unroll
    for (int t = 0; t < 4; ++t) {
        int n = nBase + t * 16 + li;
        float bias = bih[n] + bhh[n];
        #pragma unroll
        for (int r = 0; r < 8; ++r) {
            int m = mBase + r + 8 * half;
            // gx is a 268 MB/layer stream consumed once by the scan: keep it out of L2
            __builtin_nontemporal_store(acc[t][r] + bias, &G[(size_t)m * G4_ + n]);
        }
    }
}

// ---------------- bidirectional LSTM scan ----------------
// grid = 8: blockIdx.x>>2 = direction (0 fwd, 1 bwd), &3 = batch chunk of 16 rows.
// Wave w owns hidden units [32w, 32w+32): computes all 4 gate tiles for them, so the
// full LSTM cell update happens in registers; h,c persist in LDS across 512 steps.
// Whh is pre-transposed to K-major (Wt[k][n]) so B-fragment loads are coalesced.
__global__ __launch_bounds__(256)
void lstm_scan_kernel(const float* __restrict__ gxF, const float* __restrict__ gxB,
                      const float* __restrict__ WtF, const float* __restrict__ WtB,
                      const float* __restrict__ mask, float* __restrict__ out,
                      int outStride) {
    __shared__ float hS[16 * HSTR];
    __shared__ float cS[16 * HSTR];

    const int bid = blockIdx.x;
    const int dir = bid >> 2;
    const int b0  = (bid & 3) * 16;
    const float* gx = dir ? gxB : gxF;
    const float* Wt = dir ? WtB : WtF;
    const int outOff = dir ? HID_ : 0;

    const int tid = threadIdx.x;
    const int wave = tid >> 5, lane = tid & 31;
    const int li = lane & 15, half = lane >> 4;
    const int u0 = wave * 32;                        // hidden-unit base for this wave

    for (int i = tid; i < 16 * HSTR; i += 256) { hS[i] = 0.0f; cS[i] = 0.0f; }
    __syncthreads();

    // per-(gate,halftile) coalesced B column base: lanes 0-15 hit consecutive n
    const float* Bbase[4][2];
    #pragma unroll
    for (int g = 0; g < 4; ++g)
        #pragma unroll
        for (int p = 0; p < 2; ++p)
            Bbase[g][p] = Wt + (size_t)(2 * half) * G4_ + (u0 + g * HID_ + p * 16 + li);

    for (int step = 0; step < S_; ++step) {
        const int st = dir ? (S_ - 1 - step) : step;

        // prefetch this step's gx slice (16 rows x 4KB) while the GEMM runs
        {
            int mrow = tid >> 4;
            int seg  = (tid & 15) * 16;
            const float* gp = gx + ((size_t)(b0 + mrow) * S_ + st) * G4_ + seg;
            #pragma unroll
            for (int j = 0; j < 4; ++j)
                __builtin_prefetch(gp + j * 256, 0, 1);
        }

        v8f acc[4][2] = {};                          // [gate][unit-halftile]
        const float* hRow = &hS[li * HSTR + 2 * half];
        for (int k = 0; k < HID_; k += 4) {
            v2f a; a.x = hRow[k]; a.y = hRow[k + 1];
            #pragma unroll
            for (int g = 0; g < 4; ++g) {
                #pragma unroll
                for (int p = 0; p < 2; ++p) {
                    const float* Bp = Bbase[g][p] + (size_t)k * G4_;
                    v2f b; b.x = Bp[0]; b.y = Bp[G4_];
                    acc[g][p] = wmma_f32(a, b, acc[g][p]);
                }
            }
        }
        __syncthreads();   // all waves finished reading hS

        #pragma unroll
        for (int p = 0; p < 2; ++p) {
            const int n = u0 + p * 16 + li;          // hidden unit
            #pragma unroll
            for (int r = 0; r < 8; ++r) {
                const int m = r + 8 * half;          // batch row within chunk
                const int b = b0 + m;
                const size_t gbase = ((size_t)b * S_ + st) * G4_;
                float gi = acc[0][p][r] + gx[gbase + n];
                float gf = acc[1][p][r] + gx[gbase + n + HID_];
                float gg = acc[2][p][r] + gx[gbase + n + 2 * HID_];
                float go = acc[3][p][r] + gx[gbase + n + 3 * HID_];
                float i_ = fast_sigmoid(gi);
                float f_ = fast_sigmoid(gf);
                float g_ = fast_tanh(gg);
                float o_ = fast_sigmoid(go);
                float c_old = cS[m * HSTR + n];
                float c_new = f_ * c_old + i_ * g_;
                float h_new = o_ * fast_tanh(c_new);
                float mt = mask[b * S_ + st];
                float c_b = mt * c_new + (1.0f - mt) * c_old;
                float h_old = hS[m * HSTR + n];
                float h_b = mt * h_new + (1.0f - mt) * h_old;
                cS[m * HSTR + n] = c_b;
                hS[m * HSTR + n] = h_b;
                out[((size_t)b * S_ + st) * outStride + outOff + n] = mt * h_b;
            }
        }
        __syncthreads();   // hS updated before next step's GEMM
    }
}

// ---------------- emissions: (M x 512) * tagW^T(512 x 9) + tag_b ----------------
__global__ __launch_bounds__(32)
void emissions_kernel(const float* __restrict__ X, const float* __restrict__ tagW,
                      const float* __restrict__ tagb, float* __restrict__ E) {
    const int m0 = blockIdx.x * 16;
    const int lane = threadIdx.x, li = lane & 15, half = lane >> 4;
    const bool nv = (li < NL_);
    const int ns = nv ? li : (NL_ - 1);

    v8f acc = {};
    const float* Ar = X + (size_t)(m0 + li) * 512 + 2 * half;
    const float* Br = tagW + (size_t)ns * 512 + 2 * half;
    for (int k = 0; k < 512; k += 4) {
        v2f a; a.x = Ar[k]; a.y = Ar[k + 1];
        v2f b; b.x = nv ? Br[k] : 0.0f; b.y = nv ? Br[k + 1] : 0.0f;
        acc = wmma_f32(a, b, acc);
    }
    if (nv) {
        float bb = tagb[li];
        #pragma unroll
        for (int r = 0; r < 8; ++r)
            E[(size_t)(m0 + r + 8 * half) * NL_ + li] = acc[r] + bb;
    }
}

// ---------------- CRF forward + gold score, one wave per batch element ----------------
__global__ __launch_bounds__(32)
void crf_kernel(const float* __restrict__ E, const int* __restrict__ labels,
                const float* __restrict__ mask, const float* __restrict__ T,
                const float* __restrict__ startT, const float* __restrict__ endT,
                float* __restrict__ outB) {
    const int b = blockIdx.x;
    const int lane = threadIdx.x;
    const bool jv = (lane < NL_);
    const int js = jv ? lane : 0;

    float Tc[NL_];
    #pragma unroll
    for (int i = 0; i < NL_; ++i) Tc[i] = T[i * NL_ + js];

    float score = jv ? (startT[js] + E[(size_t)b * S_ * NL_ + js]) : -3.0e38f;

    for (int t = 1; t < S_; ++t) {
        float mval = -3.0e38f, ssum = 0.0f;
        #pragma unroll
        for (int i = 0; i < NL_; ++i) {
            float si = __shfl(score, i, 32);
            float v = si + Tc[i];
            float mn = fmaxf(mval, v);
            ssum = ssum * __expf(mval - mn) + __expf(v - mn);
            mval = mn;
        }
        float em = jv ? E[((size_t)b * S_ + t) * NL_ + js] : 0.0f;
        float nxt = mval + __logf(ssum) + em;
        float mt = mask[b * S_ + t];
        if (jv) score = mt * nxt + (1.0f - mt) * score;
    }

    // partition = logsumexp_j(score_j + end_j)
    float v = jv ? (score + endT[js]) : -3.0e38f;
    float m = v;
    for (int off = 16; off >= 1; off >>= 1) m = fmaxf(m, __shfl_xor(m, off, 32));
    float e = __expf(v - m);
    for (int off = 16; off >= 1; off >>= 1) e += __shfl_xor(e, off, 32);
    float partition = m + __logf(e);

    // gold score
    float acc = 0.0f;
    for (int t = 1 + lane; t < S_; t += 32) {
        int lp = labels[b * S_ + t - 1];
        int lc = labels[b * S_ + t];
        acc += (T[lp * NL_ + lc] + E[((size_t)b * S_ + t) * NL_ + lc]) * mask[b * S_ + t];
    }
    for (int off = 16; off >= 1; off >>= 1) acc += __shfl_xor(acc, off, 32);

    if (lane == 0) {
        int l0 = labels[b * S_];
        float gold = startT[l0] + E[(size_t)b * S_ * NL_ + l0] + acc;
        outB[b] = partition - gold;
    }
}

__global__ void finalize_kernel(const float* __restrict__ outB, float* __restrict__ out) {
    if (threadIdx.x == 0 && blockIdx.x == 0) {
        float s = 0.0f;
        for (int i = 0; i < B_; ++i) s += outB[i];
        out[0] = s / (float)B_;
    }
}

extern "C" void kernel_launch(void* const* d_in, const int* in_sizes, int n_in,
                              void* d_out, int out_size, void* d_ws, size_t ws_size,
                              hipStream_t stream) {
    const int*   tokens = (const int*)d_in[0];
    const int*   labels = (const int*)d_in[1];
    /* d_in[2] = lengths (unused; mask provided) */
    const float* mask   = (const float*)d_in[3];
    const float* emb    = (const float*)d_in[4];
    // lstm_params flattened: layer0(dir0: Wih,Whh,bih,bhh; dir1: ...), layer1(...)
    const float* Wih_f0 = (const float*)d_in[5];
    const float* Whh_f0 = (const float*)d_in[6];
    const float* bih_f0 = (const float*)d_in[7];
    const float* bhh_f0 = (const float*)d_in[8];
    const float* Wih_b0 = (const float*)d_in[9];
    const float* Whh_b0 = (const float*)d_in[10];
    const float* bih_b0 = (const float*)d_in[11];
    const float* bhh_b0 = (const float*)d_in[12];
    const float* Wih_f1 = (const float*)d_in[13];
    const float* Whh_f1 = (const float*)d_in[14];
    const float* bih_f1 = (const float*)d_in[15];
    const float* bhh_f1 = (const float*)d_in[16];
    const float* Wih_b1 = (const float*)d_in[17];
    const float* Whh_b1 = (const float*)d_in[18];
    const float* bih_b1 = (const float*)d_in[19];
    const float* bhh_b1 = (const float*)d_in[20];
    const float* tagW   = (const float*)d_in[21];
    const float* tagb   = (const float*)d_in[22];
    const float* trans  = (const float*)d_in[23];
    const float* startT = (const float*)d_in[24];
    const float* endT   = (const float*)d_in[25];

    float* ws = (float*)d_ws;
    float* X0    = ws;                           // M * 128
    float* XA    = X0   + (size_t)M_ * EMB_;     // M * 512
    float* X2    = XA   + (size_t)M_ * 512;      // M * 512
    float* gxF   = X2   + (size_t)M_ * 512;      // M * 1024
    float* gxB   = gxF  + (size_t)M_ * G4_;      // M * 1024
    float* em    = gxB  + (size_t)M_ * G4_;      // M * 9
    float* nllB  = em   + (size_t)M_ * NL_;      // 64
    float* WtF0  = nllB + 64;                    // 4 x (256*1024) transposed Whh
    float* WtB0  = WtF0 + (size_t)HID_ * G4_;
    float* WtF1  = WtB0 + (size_t)HID_ * G4_;
    float* WtB1  = WtF1 + (size_t)HID_ * G4_;

    dim3 ggrid(M_ / 64, G4_ / 128);
    const int tgrid = (HID_ * G4_) / 256;        // 1024 blocks

    embed_kernel<<<M_, EMB_, 0, stream>>>(tokens, emb, X0);
    transpose_kernel<<<tgrid, 256, 0, stream>>>(Whh_f0, WtF0);
    transpose_kernel<<<tgrid, 256, 0, stream>>>(Whh_b0, WtB0);
    transpose_kernel<<<tgrid, 256, 0, stream>>>(Whh_f1, WtF1);
    transpose_kernel<<<tgrid, 256, 0, stream>>>(Whh_b1, WtB1);

    // Layer 0
    gx_gemm_kernel<EMB_><<<ggrid, 256, 0, stream>>>(X0, Wih_f0, bih_f0, bhh_f0, gxF);
    gx_gemm_kernel<EMB_><<<ggrid, 256, 0, stream>>>(X0, Wih_b0, bih_b0, bhh_b0, gxB);
    lstm_scan_kernel<<<8, 256, 0, stream>>>(gxF, gxB, WtF0, WtB0, mask, XA, 512);

    // Layer 1
    gx_gemm_kernel<512><<<ggrid, 256, 0, stream>>>(XA, Wih_f1, bih_f1, bhh_f1, gxF);
    gx_gemm_kernel<512><<<ggrid, 256, 0, stream>>>(XA, Wih_b1, bih_b1, bhh_b1, gxB);
    lstm_scan_kernel<<<8, 256, 0, stream>>>(gxF, gxB, WtF1, WtB1, mask, X2, 512);

    // Emissions + CRF
    emissions_kernel<<<M_ / 16, 32, 0, stream>>>(X2, tagW, tagb, em);
    crf_kernel<<<B_, 32, 0, stream>>>(em, labels, mask, trans, startT, endT, nllB);
    finalize_kernel<<<1, 32, 0, stream>>>(nllB, (float*)d_out);
}